// AttentionLayer_24739011625905
// MI455X (gfx1250) — compile-verified
//
#include <hip/hip_runtime.h>

typedef __attribute__((ext_vector_type(2))) float v2f;
typedef __attribute__((ext_vector_type(8))) float v8f;

#define TT 200
#define DD 64
#define NEG_INF (-4294967295.0f)

// constant-lane broadcast via v_readlane (avoids ds_bpermute + dscnt waits)
__device__ __forceinline__ float readlane_f(float x, int l) {
    return __builtin_bit_cast(float,
        __builtin_amdgcn_readlane(__builtin_bit_cast(int, x), l));
}

__global__ __launch_bounds__(128) void attn_dien_kernel(
    const float* __restrict__ q, const float* __restrict__ k,
    const float* __restrict__ v, const int* __restrict__ mask,
    const float* __restrict__ W, const float* __restrict__ bias,
    float* __restrict__ out)
{
    __shared__ float m_lds[4][64];

    const int w    = threadIdx.x >> 5;   // wave in block
    const int lane = threadIdx.x & 31;
    const int half = lane >> 4;          // 0: lanes 0-15, 1: lanes 16-31
    const int lq   = lane & 15;
    const int b    = blockIdx.x * 4 + w; // batch row for this wave

    const float* qb    = q + b * DD;
    const float* kb    = k + (size_t)b * TT * DD;
    const float* vb    = v + (size_t)b * TT * DD;
    const int*   maskb = mask + b * TT;

    // ---- preamble: m_b = (W1-W2) + q_b*W3 ; c_b = q_b.(W0+W2) + bias ----
    float c_part = 0.f;
    #pragma unroll
    for (int i = 0; i < 2; ++i) {
        int e = lane + i * 32;
        float qe = qb[e];
        float u  = W[e]      + W[128 + e];
        float wv = W[64 + e] - W[128 + e];
        float se = W[192 + e];
        m_lds[w][e] = wv + qe * se;
        c_part += qe * u;
    }
    #pragma unroll
    for (int off = 16; off > 0; off >>= 1)
        c_part += __shfl_xor(c_part, off, 32);
    const float cb = c_part + bias[0];
    __syncthreads();

    // B-matrix (broadcast columns) for WMMA f32 16x16x4:
    // lanes 0-15 hold K=0,1 ; lanes 16-31 hold K=2,3  (mirrors A layout)
    v2f mB[16];
    #pragma unroll
    for (int c = 0; c < 16; ++c)
        mB[c] = *(const v2f*)&m_lds[w][4 * c + 2 * half];

    // ---- single streaming pass with online softmax ----
    float Mrun = -__builtin_inff();
    float Srun = 0.f;
    float ox = 0.f, oy = 0.f;

    for (int tile = 0; tile < 13; ++tile) {
        const int t0   = tile * 16;
        const int trow = t0 + lq;
        const int tcl  = trow < TT ? trow : TT - 1;         // clamp tail
        const float* arow = kb + tcl * DD + 2 * half;

        // scores for 16 keys: two independent 8-deep WMMA chains
        v8f acc0 = {};
        v8f acc1 = {};
        #pragma unroll
        for (int c = 0; c < 8; ++c) {
            v2f a0 = *(const v2f*)(arow + 8 * c);
            v2f a1 = *(const v2f*)(arow + 8 * c + 4);
            acc0 = __builtin_amdgcn_wmma_f32_16x16x4_f32(
                     false, a0, false, mB[2 * c],     (short)0, acc0, false, false);
            acc1 = __builtin_amdgcn_wmma_f32_16x16x4_f32(
                     false, a1, false, mB[2 * c + 1], (short)0, acc1, false, false);
        }
        v8f acc = acc0 + acc1;

        // this lane's 8 scores are t = tb .. tb+7
        const int tb   = t0 + 8 * half;
        const int tb_c = tb <= 192 ? tb : 192;              // in-bounds, 16B aligned
        const int4 m4a = *(const int4*)(maskb + tb_c);
        const int4 m4b = *(const int4*)(maskb + tb_c + 4);
        const int  mv[8] = {m4a.x, m4a.y, m4a.z, m4a.w, m4b.x, m4b.y, m4b.z, m4b.w};

        float p[8];
        float lmax = -__builtin_inff();
        #pragma unroll
        for (int r = 0; r < 8; ++r) {
            int t = tb + r;
            float sc = acc[r] + cb;
            sc = sc > 0.f ? sc : 0.f;                       // relu
            sc = (mv[r] == 0) ? NEG_INF : sc;               // padding sentinel
            sc = (t < TT) ? sc : -__builtin_inff();         // tail -> exp 0
            p[r] = sc;
            lmax = fmaxf(lmax, sc);
        }
        float tmax  = fmaxf(readlane_f(lmax, 0), readlane_f(lmax, 16));
        float Mnew  = fmaxf(Mrun, tmax);                    // always finite
        float scale = __expf(Mrun - Mnew);                  // 0 on first tile
        float lsum = 0.f;
        #pragma unroll
        for (int r = 0; r < 8; ++r) {
            p[r] = __expf(p[r] - Mnew);
            lsum += p[r];
        }
        float tsum = readlane_f(lsum, 0) + readlane_f(lsum, 16);
        Srun = Srun * scale + tsum;
        Mrun = Mnew;
        ox *= scale; oy *= scale;

        // weighted V accumulation: coalesced 256B row loads (float2 / lane),
        // weights broadcast as SGPRs via v_readlane
        const float* vrow = vb + 2 * lane;
        #pragma unroll
        for (int r = 0; r < 16; ++r) {
            int t  = t0 + r;
            int tc = t < TT ? t : TT - 1;                   // pt==0 for tail
            float pt = (r < 8) ? readlane_f(p[r], 0) : readlane_f(p[r - 8], 16);
            v2f vv = *(const v2f*)(vrow + tc * DD);
            ox += pt * vv.x;
            oy += pt * vv.y;
        }
    }

    float inv = 1.f / Srun;
    v2f res; res.x = ox * inv; res.y = oy * inv;
    *(v2f*)(out + b * DD + 2 * lane) = res;
}

extern "C" void kernel_launch(void* const* d_in, const int* in_sizes, int n_in,
                              void* d_out, int out_size, void* d_ws, size_t ws_size,
                              hipStream_t stream) {
    const float* q    = (const float*)d_in[0];
    const float* k    = (const float*)d_in[1];
    const float* v    = (const float*)d_in[2];
    const int*   mask = (const int*)d_in[3];
    const float* W    = (const float*)d_in[4];
    const float* bias = (const float*)d_in[5];
    float* out = (float*)d_out;

    const int B = in_sizes[0] / DD;      // 4096
    dim3 grid(B / 4), block(128);        // 1 wave32 per batch row
    hipLaunchKernelGGL(attn_dien_kernel, grid, block, 0, stream,
                       q, k, v, mask, W, bias, out);
}